// GCN_model_3487513445090
// MI455X (gfx1250) — compile-verified
//
#include <hip/hip_runtime.h>

#define NN 131072
#define EE 8388608
#define CC 16
#define BB 4

typedef __attribute__((ext_vector_type(2))) float v2f;
typedef __attribute__((ext_vector_type(8))) float v8f;

// ---------------------------------------------------------------------------
// Kernel 1: zero deg (ws[0..N)) and agg0 (ws[3N..4N)); seed out with fc_b.
// Must run every launch: atomics below accumulate into these buffers.
// ---------------------------------------------------------------------------
__global__ void gcn_init_kernel(float* __restrict__ ws,
                                const float* __restrict__ fc_b,
                                float* __restrict__ out) {
    int i = blockIdx.x * blockDim.x + threadIdx.x;
    if (i < NN) {
        ws[i]          = 0.0f;   // deg
        ws[3 * NN + i] = 0.0f;   // agg0
    }
    if (i < BB * CC) out[i] = fc_b[i & (CC - 1)];
}

// ---------------------------------------------------------------------------
// Kernel 2: deg[src[e]] += 1  (counts <= E < 2^24, exact in f32)
// ---------------------------------------------------------------------------
__global__ void gcn_degree_kernel(const int* __restrict__ src,
                                  float* __restrict__ deg) {
    int stride = gridDim.x * blockDim.x;
    for (int e = blockIdx.x * blockDim.x + threadIdx.x; e < EE; e += stride)
        atomicAdd(&deg[src[e]], 1.0f);
}

// ---------------------------------------------------------------------------
// Kernel 3: dscale = deg^-1/2 ; xs0 = nodes_feat[batch0] * dscale
// ---------------------------------------------------------------------------
__global__ void gcn_scale_kernel(const float* __restrict__ nodes,
                                 float* __restrict__ ws) {
    int n = blockIdx.x * blockDim.x + threadIdx.x;
    if (n < NN) {
        float ds = rsqrtf(ws[n]);        // deg >= 1 guaranteed by construction
        ws[NN + n]     = ds;             // dscale
        ws[2 * NN + n] = nodes[n] * ds;  // xs0 (batch 0 = first N floats)
    }
}

// ---------------------------------------------------------------------------
// Kernel 4: agg0[src[e]] += xs0[dst[e]]   (dominant cost: 67MB edge reads,
// 8.4M float atomics into an L2-resident 512KB array)
// ---------------------------------------------------------------------------
__global__ void gcn_scatter_kernel(const int* __restrict__ edges,
                                   const float* __restrict__ xs0,
                                   float* __restrict__ agg0) {
    const int* __restrict__ src = edges;       // edges_index_list[0,0,:]
    const int* __restrict__ dst = edges + EE;  // edges_index_list[0,1,:]
    int stride = gridDim.x * blockDim.x;
    for (int e = blockIdx.x * blockDim.x + threadIdx.x; e < EE; e += stride)
        atomicAdd(&agg0[src[e]], xs0[dst[e]]);
}

// ---------------------------------------------------------------------------
// Kernel 5: out[0,c] += sum_n fc_w[c,n] * (agg0[n]*dscale[n]) via
// V_WMMA_F32_16X16X4_F32.  A = 16 classes x 4 cols of fc_w; B = z replicated
// across all 16 N-columns (half-wave K layout respected), so every D column
// carries the same partial dot. Accumulate over the wave's n-stripe in C,
// then lanes 0 / 16 commit column 0 (M=0..7 / M=8..15) with atomics.
// ---------------------------------------------------------------------------
__global__ void gcn_fc_wmma_kernel(const float* __restrict__ fc_w,
                                   const float* __restrict__ ws,
                                   float* __restrict__ out,
                                   int numWaves) {
    const float* __restrict__ dscale = ws + NN;
    const float* __restrict__ agg0   = ws + 3 * NN;

    int tid  = blockIdx.x * blockDim.x + threadIdx.x;
    int wave = tid >> 5;
    int lane = threadIdx.x & 31;
    int row  = lane & 15;              // A-matrix row M (class index)
    int koff = (lane >> 4) << 1;       // lanes 0-15 -> K0,K1 ; lanes 16-31 -> K2,K3

    const float* __restrict__ wrow = fc_w + (size_t)row * NN;

    v8f acc = {};
    for (int n = wave * 4; n < NN; n += numWaves * 4) {   // wave-uniform: EXEC all 1s
        int idx = n + koff;
        v2f A, Bv;
        A.x  = wrow[idx];
        A.y  = wrow[idx + 1];
        Bv.x = agg0[idx]     * dscale[idx];
        Bv.y = agg0[idx + 1] * dscale[idx + 1];
        // 8 args: (neg_a, A, neg_b, B, c_mod, C, reuse_a, reuse_b)
        acc = __builtin_amdgcn_wmma_f32_16x16x4_f32(
                  false, A, false, Bv, (short)0, acc, false, false);
    }

    if (lane == 0) {
#pragma unroll
        for (int r = 0; r < 8; ++r) atomicAdd(&out[r], acc[r]);        // M = 0..7
    } else if (lane == 16) {
#pragma unroll
        for (int r = 0; r < 8; ++r) atomicAdd(&out[8 + r], acc[r]);    // M = 8..15
    }
}

// ---------------------------------------------------------------------------
extern "C" void kernel_launch(void* const* d_in, const int* in_sizes, int n_in,
                              void* d_out, int out_size, void* d_ws, size_t ws_size,
                              hipStream_t stream) {
    const float* nodes = (const float*)d_in[0];   // (B,N,1) f32
    const float* fc_w  = (const float*)d_in[1];   // (C,N)   f32
    const float* fc_b  = (const float*)d_in[2];   // (C,)    f32
    const int*   edges = (const int*)d_in[3];     // (B,2,E) i32
    // d_in[4] graph_label_list: unused by the reference output

    float* out = (float*)d_out;                   // (B,1,C) f32 = 64 elems
    float* ws  = (float*)d_ws;                    // deg | dscale | xs0 | agg0 (4*N f32)

    gcn_init_kernel   <<<(NN + 255) / 256, 256, 0, stream>>>(ws, fc_b, out);
    gcn_degree_kernel <<<2048, 256, 0, stream>>>(edges, ws);
    gcn_scale_kernel  <<<(NN + 255) / 256, 256, 0, stream>>>(nodes, ws);
    gcn_scatter_kernel<<<2048, 256, 0, stream>>>(edges, ws + 2 * NN, ws + 3 * NN);

    const int blocks = 64, threads = 256;
    const int numWaves = blocks * threads / 32;   // 512 waves -> 64 WMMA iters each
    gcn_fc_wmma_kernel<<<blocks, threads, 0, stream>>>(fc_w, ws, out, numWaves);
}